// HGCNLayer_42236708388941
// MI455X (gfx1250) — compile-verified
//
#include <hip/hip_runtime.h>
#include <hip/hip_bf16.h>
#include <math.h>

// ---------------------------------------------------------------------------
// Problem constants (match reference)
// ---------------------------------------------------------------------------
#define NN     6144
#define DIN    512
#define DOUT   512

typedef __attribute__((ext_vector_type(2))) float v2f;
typedef __attribute__((ext_vector_type(8))) float v8f;

// ---------------------------------------------------------------------------
// Kernel 1: fp32 WMMA GEMM  C[M=6144, N=512] = A[6144,512] x B[512,512]
// blockIdx.z == 0 -> B = W_sa   -> C = h
// blockIdx.z == 1 -> B = W_gcnb -> C = g
// Tile: 128(M) x 64(N) per 256-thread block (8 waves, each 16x64),
// K staged through LDS in chunks of 64, WMMA shape 16x16x4 (pure fp32).
// B is stored TRANSPOSED in LDS (Bt[n][k]) so each B fragment is one
// aligned ds_load_b64; stride 68 makes both A and B reads bank-conflict-free.
// ---------------------------------------------------------------------------
#define TILE_M 128
#define TILE_N 64
#define KCHUNK 64
#define LDS_STRIDE 68   // 64 + 4 pad

__global__ __launch_bounds__(256)
void gemm_f32_wmma(const float* __restrict__ x,
                   const float* __restrict__ Wsa,
                   const float* __restrict__ Wgcn,
                   float* __restrict__ h,
                   float* __restrict__ g)
{
    __shared__ float As[TILE_M][LDS_STRIDE];   // As[m][k]
    __shared__ float Bt[TILE_N][LDS_STRIDE];   // Bt[n][k]  (transposed)

    const float* __restrict__ B = (blockIdx.z == 0) ? Wsa : Wgcn;
    float* __restrict__ C       = (blockIdx.z == 0) ? h   : g;

    const int n0  = blockIdx.x * TILE_N;
    const int m0  = blockIdx.y * TILE_M;
    const int tid = threadIdx.x;
    const int lane = tid & 31;
    const int wave = tid >> 5;       // 0..7 -> M sub-tile
    const int l16  = lane & 15;
    const int hi   = lane >> 4;      // 0 or 1

    v8f acc[4];
    #pragma unroll
    for (int t = 0; t < 4; ++t) acc[t] = (v8f)(0.0f);

    for (int k0 = 0; k0 < DIN; k0 += KCHUNK) {
        // ---- stage A tile: 128 x 64 floats = 2048 float4, 8 per thread ----
        #pragma unroll
        for (int it = 0; it < 8; ++it) {
            int idx = it * 256 + tid;            // 0..2047
            int r   = idx >> 4;                  // 0..127 (M)
            int c4  = idx & 15;                  // float4 column (K)
            float4 v = *(const float4*)(x + (size_t)(m0 + r) * DIN + k0 + c4 * 4);
            As[r][c4 * 4 + 0] = v.x;
            As[r][c4 * 4 + 1] = v.y;
            As[r][c4 * 4 + 2] = v.z;
            As[r][c4 * 4 + 3] = v.w;
        }
        // ---- stage B tile transposed: read rows (coalesced), scatter to Bt ----
        #pragma unroll
        for (int it = 0; it < 4; ++it) {
            int idx = it * 256 + tid;            // 0..1023
            int r   = idx >> 4;                  // 0..63 (K row)
            int c4  = idx & 15;                  // float4 col (N)
            float4 v = *(const float4*)(B + (size_t)(k0 + r) * DOUT + n0 + c4 * 4);
            Bt[c4 * 4 + 0][r] = v.x;
            Bt[c4 * 4 + 1][r] = v.y;
            Bt[c4 * 4 + 2][r] = v.z;
            Bt[c4 * 4 + 3][r] = v.w;
        }
        __syncthreads();

        // ---- 16 K-steps of 4, 4 N sub-tiles each -> 64 WMMAs per chunk ----
        #pragma unroll
        for (int kk = 0; kk < KCHUNK; kk += 4) {
            // lanes 0-15 hold K=kk,kk+1 ; lanes 16-31 hold kk+2,kk+3
            const int ak = kk + hi * 2;
            float2 av = *(const float2*)&As[wave * 16 + l16][ak];
            v2f a; a.x = av.x; a.y = av.y;
            #pragma unroll
            for (int t = 0; t < 4; ++t) {
                float2 bv = *(const float2*)&Bt[t * 16 + l16][ak];
                v2f b; b.x = bv.x; b.y = bv.y;
                acc[t] = __builtin_amdgcn_wmma_f32_16x16x4_f32(
                    false, a, false, b, (short)0, acc[t], false, false);
            }
        }
        __syncthreads();
    }

    // ---- store: C layout VGPR v -> rows (v) [lanes 0-15] and (v+8) [16-31] ----
    #pragma unroll
    for (int t = 0; t < 4; ++t) {
        #pragma unroll
        for (int v = 0; v < 8; ++v) {
            int row = m0 + wave * 16 + v + hi * 8;
            int col = n0 + t * 16 + l16;
            C[(size_t)row * DOUT + col] = acc[t][v];
        }
    }
}

// ---------------------------------------------------------------------------
// Kernel 2: per-row attention scores  s_left = h@a[:512], s_right = h@a[512:]
// one wave (32 lanes) per row, 8 rows per 256-thread block
// ---------------------------------------------------------------------------
__global__ __launch_bounds__(256)
void scores_kernel(const float* __restrict__ h,
                   const float* __restrict__ a_sa,
                   float* __restrict__ sl,
                   float* __restrict__ sr)
{
    const int lane = threadIdx.x & 31;
    const int wave = threadIdx.x >> 5;
    const int row  = blockIdx.x * 8 + wave;
    const float* hp = h + (size_t)row * DOUT;
    float l = 0.f, r = 0.f;
    for (int k = lane; k < DOUT; k += 32) {
        float hv = hp[k];
        l += hv * a_sa[k];
        r += hv * a_sa[DOUT + k];
    }
    #pragma unroll
    for (int off = 16; off; off >>= 1) {
        l += __shfl_xor(l, off, 32);
        r += __shfl_xor(r, off, 32);
    }
    if (lane == 0) { sl[row] = l; sr[row] = r; }
}

// ---------------------------------------------------------------------------
// block-wide sum (256 threads, 8 waves). Self-synchronizing around `red`.
// ---------------------------------------------------------------------------
__device__ __forceinline__ float block_sum(float v, float* red)
{
    #pragma unroll
    for (int off = 16; off; off >>= 1) v += __shfl_xor(v, off, 32);
    const int lane = threadIdx.x & 31;
    const int wave = threadIdx.x >> 5;
    __syncthreads();                 // protect red from any previous use
    if (lane == 0) red[wave] = v;
    __syncthreads();
    float s = 0.f;
    #pragma unroll
    for (int w = 0; w < 8; ++w) s += red[w];
    return s;
}

__device__ __forceinline__ float sigmoidf_(float z)
{
    return 1.0f / (1.0f + __expf(-z));
}

// Adjacency scan chunk: 256 threads x float4 = 1024 columns per iteration.
#define CHUNK_COLS 1024

// ---------------------------------------------------------------------------
// Kernel 3: fused sparse pass over adj_a (one 256-thread block per row i):
//   - float4 stream of the row + prefetch of next chunk (global_prefetch_b8)
//   - compact nonzeros into LDS (ds_add_rtn_u32)
//   - w_ij = exp(-leaky_relu(sl[i]+sr[j], 0.01)), denom = sum w
//   - x_a[i,:]  = (sum_j w_ij * h[j,:]) / (denom + 1e-5)
//   - attn_a[i] = sigmoid( dot(Wa[:512], sum_j x[j,:]) + dot(Wa[512:], x[i,:]) + ba )
// Each thread owns output dims {2t, 2t+1}.
// ---------------------------------------------------------------------------
__global__ __launch_bounds__(256)
void pass_a_kernel(const float* __restrict__ adj_a,
                   const float* __restrict__ x,
                   const float* __restrict__ h,
                   const float* __restrict__ sl,
                   const float* __restrict__ sr,
                   const float* __restrict__ Wa,
                   const float* __restrict__ ba,
                   float* __restrict__ x_a,
                   float* __restrict__ attn_a)
{
    const int i   = blockIdx.x;
    const int tid = threadIdx.x;
    __shared__ int   jlist[CHUNK_COLS];
    __shared__ float wlist[CHUNK_COLS];
    __shared__ int   cnt;
    __shared__ float red[8];

    float ts0 = 0.f, ts1 = 0.f;      // type_sum_a dims 2t,2t+1
    float xa0 = 0.f, xa1 = 0.f;      // unnormalized x_a dims
    float den = 0.f;
    const float sli = sl[i];
    const float* __restrict__ arow = adj_a + (size_t)i * NN;

    for (int j0 = 0; j0 < NN; j0 += CHUNK_COLS) {
        if (tid == 0) cnt = 0;
        __syncthreads();
        const float4 av = *(const float4*)(arow + j0 + 4 * tid);
        if (j0 + CHUNK_COLS < NN)
            __builtin_prefetch(arow + j0 + CHUNK_COLS + 4 * tid, 0, 1);
        const float vals[4] = { av.x, av.y, av.z, av.w };
        #pragma unroll
        for (int q = 0; q < 4; ++q) {
            if (vals[q] != 0.0f) {
                const int j = j0 + 4 * tid + q;
                float s  = sli + sr[j];
                float lr = (s > 0.f) ? s : 0.01f * s;
                float w  = __expf(-lr);
                int p = atomicAdd(&cnt, 1);     // LDS compaction
                jlist[p] = j;
                wlist[p] = w;
                den += w;
            }
        }
        __syncthreads();
        const int c = cnt;
        for (int k = 0; k < c; ++k) {
            const int   j = jlist[k];
            const float w = wlist[k];
            const float2 xv = *(const float2*)(x + (size_t)j * DIN  + 2 * tid);
            const float2 hv = *(const float2*)(h + (size_t)j * DOUT + 2 * tid);
            ts0 += xv.x;        ts1 += xv.y;
            xa0 += w * hv.x;    xa1 += w * hv.y;
        }
        __syncthreads();
    }

    const float denom = block_sum(den, red) + 1e-5f;
    const float inv   = 1.0f / denom;

    // type-level gate
    const float2 xi = *(const float2*)(x + (size_t)i * DIN + 2 * tid);
    float pa = Wa[2 * tid] * ts0 + Wa[2 * tid + 1] * ts1
             + Wa[DIN + 2 * tid] * xi.x + Wa[DIN + 2 * tid + 1] * xi.y;
    const float dv = block_sum(pa, red);
    const float ga = sigmoidf_(dv + ba[0]);

    float2 o; o.x = xa0 * inv; o.y = xa1 * inv;
    *(float2*)(x_a + (size_t)i * DOUT + 2 * tid) = o;
    if (tid == 0) attn_a[i] = ga;
}

// ---------------------------------------------------------------------------
// Kernel 4: fused sparse pass over adj_b + final combine:
//   x_b[i,:]  = sum_j g[j,:] + b_gcnb          (g = x @ W_gcnb)
//   attn_b[i] = sigmoid(dot(Wb[:512], sum_j x[j,:]) + dot(Wb[512:], x[i,:]) + bb)
//   out[i,:]  = sigmoid(attn_a[i]*x_a[i,:] + attn_b[i]*x_b[i,:])
// ---------------------------------------------------------------------------
__global__ __launch_bounds__(256)
void pass_b_final_kernel(const float* __restrict__ adj_b,
                         const float* __restrict__ x,
                         const float* __restrict__ g,
                         const float* __restrict__ b_gcnb,
                         const float* __restrict__ Wb,
                         const float* __restrict__ bb,
                         const float* __restrict__ x_a,
                         const float* __restrict__ attn_a,
                         float* __restrict__ out)
{
    const int i   = blockIdx.x;
    const int tid = threadIdx.x;
    __shared__ int   jlist[CHUNK_COLS];
    __shared__ int   cnt;
    __shared__ float red[8];

    float ts0 = 0.f, ts1 = 0.f;
    float xb0 = 0.f, xb1 = 0.f;
    const float* __restrict__ brow = adj_b + (size_t)i * NN;

    for (int j0 = 0; j0 < NN; j0 += CHUNK_COLS) {
        if (tid == 0) cnt = 0;
        __syncthreads();
        const float4 av = *(const float4*)(brow + j0 + 4 * tid);
        if (j0 + CHUNK_COLS < NN)
            __builtin_prefetch(brow + j0 + CHUNK_COLS + 4 * tid, 0, 1);
        const float vals[4] = { av.x, av.y, av.z, av.w };
        #pragma unroll
        for (int q = 0; q < 4; ++q) {
            if (vals[q] != 0.0f) {
                int p = atomicAdd(&cnt, 1);
                jlist[p] = j0 + 4 * tid + q;
            }
        }
        __syncthreads();
        const int c = cnt;
        for (int k = 0; k < c; ++k) {
            const int j = jlist[k];
            const float2 xv = *(const float2*)(x + (size_t)j * DIN  + 2 * tid);
            const float2 gv = *(const float2*)(g + (size_t)j * DOUT + 2 * tid);
            ts0 += xv.x;  ts1 += xv.y;
            xb0 += gv.x;  xb1 += gv.y;
        }
        __syncthreads();
    }

    xb0 += b_gcnb[2 * tid];
    xb1 += b_gcnb[2 * tid + 1];

    const float2 xi = *(const float2*)(x + (size_t)i * DIN + 2 * tid);
    float pb = Wb[2 * tid] * ts0 + Wb[2 * tid + 1] * ts1
             + Wb[DIN + 2 * tid] * xi.x + Wb[DIN + 2 * tid + 1] * xi.y;
    const float dv = block_sum(pb, red);
    const float gb = sigmoidf_(dv + bb[0]);
    const float ga = attn_a[i];

    const float2 xav = *(const float2*)(x_a + (size_t)i * DOUT + 2 * tid);
    float2 o;
    o.x = sigmoidf_(ga * xav.x + gb * xb0);
    o.y = sigmoidf_(ga * xav.y + gb * xb1);
    *(float2*)(out + (size_t)i * DOUT + 2 * tid) = o;
}

// ---------------------------------------------------------------------------
// Launch
// ---------------------------------------------------------------------------
extern "C" void kernel_launch(void* const* d_in, const int* in_sizes, int n_in,
                              void* d_out, int out_size, void* d_ws, size_t ws_size,
                              hipStream_t stream)
{
    const float* x      = (const float*)d_in[0];
    const float* adj_a  = (const float*)d_in[1];
    const float* adj_b  = (const float*)d_in[2];
    const float* W_sa   = (const float*)d_in[3];
    const float* a_sa   = (const float*)d_in[4];
    const float* W_gcnb = (const float*)d_in[5];
    const float* b_gcnb = (const float*)d_in[6];
    const float* Wa     = (const float*)d_in[7];
    const float* ba     = (const float*)d_in[8];
    const float* Wb     = (const float*)d_in[9];
    const float* bb     = (const float*)d_in[10];
    float* out = (float*)d_out;

    // workspace layout (floats): h, g, x_a  (3 x 12.6MB)  +  sl, sr, attn_a
    float* ws   = (float*)d_ws;
    float* h    = ws;
    float* g    = h  + (size_t)NN * DOUT;
    float* xa   = g  + (size_t)NN * DOUT;
    float* sl   = xa + (size_t)NN * DOUT;
    float* sr   = sl + NN;
    float* ata  = sr + NN;

    // 1) dual fp32-WMMA GEMM: h = x@W_sa, g = x@W_gcnb
    dim3 ggrid(DOUT / TILE_N, NN / TILE_M, 2);   // (8, 48, 2)
    gemm_f32_wmma<<<ggrid, 256, 0, stream>>>(x, W_sa, W_gcnb, h, g);

    // 2) per-row attention score halves
    scores_kernel<<<NN / 8, 256, 0, stream>>>(h, a_sa, sl, sr);

    // 3) fused sparse attention pass over adj_a
    pass_a_kernel<<<NN, 256, 0, stream>>>(adj_a, x, h, sl, sr, Wa, ba, xa, ata);

    // 4) fused GCN pass over adj_b + final gated sigmoid combine
    pass_b_final_kernel<<<NN, 256, 0, stream>>>(adj_b, x, g, b_gcnb, Wb, bb,
                                                xa, ata, out);
}